// HierarchicalSoftmaxEnsemble_34248069219028
// MI455X (gfx1250) — compile-verified
//
#include <hip/hip_runtime.h>
#include <hip/hip_bf16.h>

// Hierarchical-softmax ensemble: gather-GEMV instead of dense GEMM.
//   x:[32,1024] f32, W:[131068,1024] f32, b:[131068] f32,
//   comb_idx:[32768,60] i32, comb_labels:[32768,60] f32, target:[32] i32
// out[0..1919]    = (x[b] . W[comb_idx[target[b],j]]) + bias[...]   (b-major)
// out[1920..3839] = comb_labels[target[b], j]                       (b-major)

typedef __attribute__((ext_vector_type(2))) float v2f;
typedef __attribute__((ext_vector_type(8))) float v8f;

#define HS_B     32
#define HS_K     60      // M*D = 4*15 path positions per sample
#define HS_IN    1024
#define HS_TILES 4       // ceil(60/16) 16-row WMMA tiles per sample

__device__ __forceinline__ v8f wmma_f32_acc(v2f a, v2f b, v8f c) {
  // D = A(16x4,f32) * B(4x16,f32) + C(16x16,f32)
  return __builtin_amdgcn_wmma_f32_16x16x4_f32(
      /*neg_a=*/false, a, /*neg_b=*/false, b,
      /*c_mod=*/(short)0, c, /*reuse_a=*/false, /*reuse_b=*/false);
}

__global__ __launch_bounds__(256)
void HierarchicalSoftmaxEnsemble_kernel(const float* __restrict__ x,
                                        const float* __restrict__ W,
                                        const float* __restrict__ bias,
                                        const int*   __restrict__ comb_idx,
                                        const float* __restrict__ comb_labels,
                                        const int*   __restrict__ target,
                                        float*       __restrict__ out)
{
  const int lane   = threadIdx.x & 31;
  const int waveId = blockIdx.x * (blockDim.x >> 5) + (threadIdx.x >> 5);
  const int b      = waveId >> 2;   // sample index
  const int tile   = waveId & 3;    // which 16 of the 60 gathered rows
  if (b >= HS_B) return;            // uniform per-wave, EXEC stays all-1s

  const int t = target[b];

  // A-matrix 16x4 f32 layout: lane L holds row M = L%16, K-pair (L/16)*2 .. +1.
  const int m    = lane & 15;
  const int j    = tile * 16 + m;                       // gathered-row slot
  const int row  = (j < HS_K) ? comb_idx[t * HS_K + j] : 0;  // pad with row 0
  const int ksub = (lane >> 4) * 2;

  const float* wrow = W + (size_t)row * HS_IN + ksub;   // per-lane gathered row
  const float* xrow = x + (size_t)b  * HS_IN + ksub;    // B broadcast across N

  // 4 independent accumulators: 4 wmmas per 16-float K step, no C-chain stall.
  v8f a0 = {}, a1 = {}, a2 = {}, a3 = {};
  for (int k0 = 0; k0 < HS_IN; k0 += 16) {
    v2f wa0 = *(const v2f*)(wrow + k0 + 0);
    v2f xb0 = *(const v2f*)(xrow + k0 + 0);
    v2f wa1 = *(const v2f*)(wrow + k0 + 4);
    v2f xb1 = *(const v2f*)(xrow + k0 + 4);
    v2f wa2 = *(const v2f*)(wrow + k0 + 8);
    v2f xb2 = *(const v2f*)(xrow + k0 + 8);
    v2f wa3 = *(const v2f*)(wrow + k0 + 12);
    v2f xb3 = *(const v2f*)(xrow + k0 + 12);
    a0 = wmma_f32_acc(wa0, xb0, a0);
    a1 = wmma_f32_acc(wa1, xb1, a1);
    a2 = wmma_f32_acc(wa2, xb2, a2);
    a3 = wmma_f32_acc(wa3, xb3, a3);
  }
  v8f s = a0 + a1 + a2 + a3;   // D columns are replicated; elementwise sum ok

  // C/D layout: VGPR r = M=r (lanes 0-15) / M=8+r (lanes 16-31), all N equal.
  // Lanes 0-7 emit M=0..7, lanes 16-23 emit M=8..15.
  const bool writer = ((lane & 15) < 8);
  const int  mw = ((lane >> 4) << 3) | (lane & 7);
  const int  jw = tile * 16 + mw;
  if (writer && jw < HS_K) {
    const int roww = comb_idx[t * HS_K + jw];
    out[b * HS_K + jw] = s[lane & 7] + bias[roww];
  }

  // Second tuple output: gathered path labels.
  if (lane < 16) {
    const int jl = tile * 16 + lane;
    if (jl < HS_K)
      out[HS_B * HS_K + b * HS_K + jl] = comb_labels[t * HS_K + jl];
  }
}

extern "C" void kernel_launch(void* const* d_in, const int* in_sizes, int n_in,
                              void* d_out, int out_size, void* d_ws, size_t ws_size,
                              hipStream_t stream) {
  (void)in_sizes; (void)n_in; (void)d_ws; (void)ws_size; (void)out_size;
  const float* x           = (const float*)d_in[0];
  const float* W           = (const float*)d_in[1];
  const float* bias        = (const float*)d_in[2];
  const int*   comb_idx    = (const int*)  d_in[3];
  const float* comb_labels = (const float*)d_in[4];
  const int*   target      = (const int*)  d_in[5];
  float*       out         = (float*)d_out;

  // 32 samples x 4 tiles = 128 waves; 8 waves (256 threads) per block.
  const int totalWaves = HS_B * HS_TILES;
  const int wavesPerBlock = 8;
  dim3 grid((totalWaves + wavesPerBlock - 1) / wavesPerBlock);
  dim3 block(wavesPerBlock * 32);
  hipLaunchKernelGGL(HierarchicalSoftmaxEnsemble_kernel, grid, block, 0, stream,
                     x, W, bias, comb_idx, comb_labels, target, out);
}